// CCALayer3D_79886391705784
// MI455X (gfx1250) — compile-verified
//
#include <hip/hip_runtime.h>

typedef float v2f __attribute__((ext_vector_type(2)));
typedef float v4f __attribute__((ext_vector_type(4)));
typedef float v8f __attribute__((ext_vector_type(8)));

#define BATCH   4
#define NCH     64
#define NPLANE  (BATCH * NCH)        // 256 (b,c) planes
#define SPATIAL 524288               // 32*128*128 elements per plane
#define SPLITS  32                   // chunks per plane
#define CHUNK4  4096                 // float4 per (plane,split) chunk = 16384 floats
#define ITERS   16                   // float4 per thread (256 threads * 16 = 4096)

// ---------------------------------------------------------------------------
// Pass 1: per-(b,c) partial sum / sum-of-squares. Deterministic tree reduce.
// ---------------------------------------------------------------------------
__global__ __launch_bounds__(256) void cca_reduce(const float* __restrict__ x,
                                                  float* __restrict__ psum,
                                                  float* __restrict__ psq) {
  const int s     = blockIdx.x;
  const int plane = blockIdx.y;
  const v4f* x4 = (const v4f*)x;
  const long base = (long)plane * (SPATIAL / 4) + (long)s * CHUNK4;

  float sum = 0.f, sq = 0.f;
#pragma unroll
  for (int i = 0; i < ITERS; ++i) {
    v4f v = x4[base + i * 256 + threadIdx.x];
    sum += (v.x + v.y) + (v.z + v.w);
    sq = fmaf(v.x, v.x, sq);
    sq = fmaf(v.y, v.y, sq);
    sq = fmaf(v.z, v.z, sq);
    sq = fmaf(v.w, v.w, sq);
  }
  // wave32 butterfly reduce
#pragma unroll
  for (int off = 16; off >= 1; off >>= 1) {
    sum += __shfl_xor(sum, off);
    sq  += __shfl_xor(sq,  off);
  }
  __shared__ float wsum[8], wsq[8];
  const int wave = threadIdx.x >> 5;
  if ((threadIdx.x & 31) == 0) { wsum[wave] = sum; wsq[wave] = sq; }
  __syncthreads();
  if (threadIdx.x == 0) {
    float S = 0.f, Q = 0.f;
#pragma unroll
    for (int w = 0; w < 8; ++w) { S += wsum[w]; Q += wsq[w]; }
    psum[plane * SPLITS + s] = S;
    psq [plane * SPLITS + s] = Q;
  }
}

// ---------------------------------------------------------------------------
// Pass 2: finish stats, then squeeze-excite MLP on one wave via
// V_WMMA_F32_16X16X4_F32 (f32 WMMA keeps full precision for g).
//   FC1: h[b][r] = relu( sum_c y[b][c] * w1[r][c] + b1[r] )   (K=64 -> 16 WMMAs)
//   FC2: g[b][c] = sigmoid( sum_r h[b][r] * w2[c][r] + b2[c] ) (4 column tiles)
// ---------------------------------------------------------------------------
__global__ __launch_bounds__(256) void cca_fc(const float* __restrict__ psum,
                                              const float* __restrict__ psq,
                                              const float* __restrict__ w1,
                                              const float* __restrict__ b1,
                                              const float* __restrict__ w2,
                                              const float* __restrict__ b2,
                                              float* __restrict__ g) {
  __shared__ float yls[NPLANE];
  {
    const int p = threadIdx.x;   // 256 threads <-> 256 planes
    float S = 0.f, Q = 0.f;
#pragma unroll 4
    for (int i = 0; i < SPLITS; ++i) {
      S += psum[p * SPLITS + i];
      Q += psq [p * SPLITS + i];
    }
    const float invN = 1.f / (float)SPATIAL;
    const float mean = S * invN;
    const float var  = fmaf(-mean, mean, Q * invN);   // E[x^2] - mean^2 (biased)
    yls[p] = sqrtf(fmaxf(var, 0.f)) + mean;           // std + mean
  }
  __syncthreads();

  if (threadIdx.x < 32) {              // single full wave: EXEC all-1s for WMMA
    const int  L  = threadIdx.x;
    const int  n  = L & 15;            // row (A) / column (B) index
    const bool hi = (L >= 16);
    const int  kh = hi ? 2 : 0;        // K sub-pair held by this half-wave

    // ---- FC1: D = Ypad(16x64) * W1^T pad(64x16), 16 chained 16x16x4 WMMAs
    v8f acc = {0.f, 0.f, 0.f, 0.f, 0.f, 0.f, 0.f, 0.f};
#pragma unroll
    for (int t = 0; t < 16; ++t) {
      const int k = 4 * t + kh;
      v2f a, b;
      // A fragment: lane (n,hi) holds {Y[n][k], Y[n][k+1]} (rows 4..15 zero-pad)
      a.x = (n < 4) ? yls[n * NCH + k]     : 0.f;
      a.y = (n < 4) ? yls[n * NCH + k + 1] : 0.f;
      // B fragment: B[k][n] = w1[n][k]    (cols 4..15 zero-pad)
      b.x = (n < 4) ? w1[n * NCH + k]      : 0.f;
      b.y = (n < 4) ? w1[n * NCH + k + 1]  : 0.f;
      acc = __builtin_amdgcn_wmma_f32_16x16x4_f32(false, a, false, b,
                                                  (short)0, acc, false, false);
    }

    // D[m][r] (m,r in 0..3) lives in acc[m], lane r. Broadcast full 4x4 H.
    const float b1_0 = b1[0], b1_1 = b1[1], b1_2 = b1[2], b1_3 = b1[3];
#define HV(mm, rr) fmaxf(__shfl(acc[mm], rr) + b1_##rr, 0.f)
    const float h00 = HV(0,0), h01 = HV(0,1), h02 = HV(0,2), h03 = HV(0,3);
    const float h10 = HV(1,0), h11 = HV(1,1), h12 = HV(1,2), h13 = HV(1,3);
    const float h20 = HV(2,0), h21 = HV(2,1), h22 = HV(2,2), h23 = HV(2,3);
    const float h30 = HV(3,0), h31 = HV(3,1), h32 = HV(3,2), h33 = HV(3,3);
#undef HV

    // ---- FC2 A fragment: lane (n,hi) holds {H[n][kh], H[n][kh+1]}
    const float a0 = hi ? (n==0?h02:n==1?h12:n==2?h22:n==3?h32:0.f)
                        : (n==0?h00:n==1?h10:n==2?h20:n==3?h30:0.f);
    const float a1 = hi ? (n==0?h03:n==1?h13:n==2?h23:n==3?h33:0.f)
                        : (n==0?h01:n==1?h11:n==2?h21:n==3?h31:0.f);
    v2f a2; a2.x = a0; a2.y = a1;

#pragma unroll
    for (int T = 0; T < 4; ++T) {
      const int c = T * 16 + n;                 // output channel of this lane/tile
      v2f bf;                                   // B[k][n] = w2[c][k], w2 row-major [64,4]
      bf.x = w2[c * 4 + kh];
      bf.y = w2[c * 4 + kh + 1];
      v8f acc2 = {0.f, 0.f, 0.f, 0.f, 0.f, 0.f, 0.f, 0.f};
      acc2 = __builtin_amdgcn_wmma_f32_16x16x4_f32(false, a2, false, bf,
                                                   (short)0, acc2, false, false);
      if (!hi) {                                // rows b=0..3 in acc2[0..3], lanes 0..15
        const float bb = b2[c];
        g[0 * NCH + c] = 1.f / (1.f + __expf(-(acc2[0] + bb)));
        g[1 * NCH + c] = 1.f / (1.f + __expf(-(acc2[1] + bb)));
        g[2 * NCH + c] = 1.f / (1.f + __expf(-(acc2[2] + bb)));
        g[3 * NCH + c] = 1.f / (1.f + __expf(-(acc2[3] + bb)));
      }
    }
  }
}

// ---------------------------------------------------------------------------
// Pass 3: out = x * g[plane]. Reverse traversal so the tail of x (still hot in
// the 192MB L2 from pass 1) is consumed first. NT load (x is dead after) and
// NT store (out is never re-read) keep L2 for useful lines.
// ---------------------------------------------------------------------------
__global__ __launch_bounds__(256) void cca_scale(const float* __restrict__ x,
                                                 const float* __restrict__ g,
                                                 float* __restrict__ out) {
  const int s     = (SPLITS - 1) - blockIdx.x;
  const int plane = (NPLANE - 1) - blockIdx.y;
  const float gv = g[plane];
  const v4f* x4 = (const v4f*)x;
  v4f* o4 = (v4f*)out;
  const long base = (long)plane * (SPATIAL / 4) + (long)s * CHUNK4;
#pragma unroll
  for (int i = 0; i < ITERS; ++i) {
    const long idx = base + i * 256 + threadIdx.x;
    v4f v = __builtin_nontemporal_load(&x4[idx]);
    v.x *= gv; v.y *= gv; v.z *= gv; v.w *= gv;
    __builtin_nontemporal_store(v, &o4[idx]);
  }
}

// ---------------------------------------------------------------------------
extern "C" void kernel_launch(void* const* d_in, const int* in_sizes, int n_in,
                              void* d_out, int out_size, void* d_ws, size_t ws_size,
                              hipStream_t stream) {
  const float* x  = (const float*)d_in[0];
  const float* w1 = (const float*)d_in[1];
  const float* b1 = (const float*)d_in[2];
  const float* w2 = (const float*)d_in[3];
  const float* b2 = (const float*)d_in[4];
  float* out = (float*)d_out;

  float* ws   = (float*)d_ws;
  float* psum = ws;                           // [256*32]
  float* psq  = ws + NPLANE * SPLITS;         // [256*32]
  float* g    = ws + 2 * NPLANE * SPLITS;     // [256]

  dim3 grid(SPLITS, NPLANE);                  // 8192 blocks: plenty of MLP for HBM
  cca_reduce<<<grid, 256, 0, stream>>>(x, psum, psq);
  cca_fc<<<1, 256, 0, stream>>>(psum, psq, w1, b1, w2, b2, g);
  cca_scale<<<grid, 256, 0, stream>>>(x, g, out);
}